// FCCapsMatrixNorecept_70617852280858
// MI455X (gfx1250) — compile-verified
//
#include <hip/hip_runtime.h>
#include <math.h>

// ---------------------------------------------------------------------------
// Matrix-capsule EM routing, fused for MI455X (gfx1250, wave32).
// Votes are recomputed on the fly with V_WMMA_F32_16X16X4_F32 (K=4 pose
// matmul is the native shape); the 268MB vote tensor is never materialized.
// Workspace: R (B,O,N) 16MB + Rs/mu/sigma/a_out (~300KB). All L2-resident.
// ---------------------------------------------------------------------------

#define B_   32
#define HW_  64      // Hh*Ww
#define I_   32
#define O_   64
#define N_   2048    // HW_*I_
#define P_   16
#define EPSF 1e-8f

typedef float v2f __attribute__((ext_vector_type(2)));
typedef float v8f __attribute__((ext_vector_type(8)));

__device__ __forceinline__ v8f wmma_f32_16x16x4(v2f a, v2f b, v8f c) {
  // (neg_a, A, neg_b, B, c_mod, C, reuse_a, reuse_b)
  return __builtin_amdgcn_wmma_f32_16x16x4_f32(false, a, false, b, (short)0, c,
                                               false, false);
}

__device__ __forceinline__ void prefetch_g(const void* p) {
  __builtin_prefetch(p, 0, 3);   // lowers to global_prefetch_b8 on gfx1250
}

// ---------------------------------------------------------------------------
// R[b,o,n] = 1/O
__global__ __launch_bounds__(256) void initR_kernel(float* __restrict__ R) {
  int idx = blockIdx.x * 256 + threadIdx.x;
  R[idx] = 1.0f / (float)O_;
}

// ---------------------------------------------------------------------------
// Rs[b,o] = sum_n R[b,o,n] * a[b,n]
__global__ __launch_bounds__(256) void rs_kernel(const float* __restrict__ R,
                                                 const float* __restrict__ a,
                                                 float* __restrict__ Rs) {
  __shared__ float red[8];
  int tid = threadIdx.x;
  int bo  = blockIdx.x;       // b*O + o
  int b   = bo >> 6;
  float s = 0.f;
  for (int n = tid; n < N_; n += 256)
    s += R[(size_t)bo * N_ + n] * a[b * N_ + n];
  for (int off = 1; off < 32; off <<= 1) s += __shfl_xor(s, off, 32);
  if ((tid & 31) == 0) red[tid >> 5] = s;
  __syncthreads();
  if (tid < 8) {
    float t = red[tid];
    for (int off = 1; off < 8; off <<= 1) t += __shfl_xor(t, off, 32);
    if (tid == 0) Rs[bo] = t;
  }
}

// ---------------------------------------------------------------------------
// M-step: per block = (b, group of 16 output caps).
// Streams all n = (hw, i), recomputing votes with WMMA, accumulating
// S1 = sum coeff*V and S2 = sum coeff*V^2 (coeff = R*a/(Rs+eps)).
// Then mu = S1, sigma = S2 - S1^2*(2 - Rs/(Rs+eps)) + eps, cost, a_out.
__global__ __launch_bounds__(256) void mstep_kernel(
    const float* __restrict__ x,      // (B,HW,I,16)
    const float* __restrict__ a,      // (B,N)
    const float* __restrict__ Wm,     // (I,O,16)
    const float* __restrict__ beta_u, // (O)
    const float* __restrict__ beta_a, // (O)
    const float* __restrict__ R,      // (B,O,N)
    const float* __restrict__ Rs,     // (B,O)
    float* __restrict__ mu,           // (B,O,16)
    float* __restrict__ sigma,        // (B,O,16)
    float* __restrict__ aout,         // (B,O)
    float* __restrict__ out_pose,     // (B,O,16)
    float* __restrict__ out_act,      // (B,O)
    float lam, int write_out) {
  __shared__ float S1[256];
  __shared__ float S2[256];
  int tid  = threadIdx.x;
  int lane = tid & 31;
  int wave = tid >> 5;
  int b    = blockIdx.x >> 2;
  int og   = blockIdx.x & 3;
  int o0   = og * 16;

  S1[tid] = 0.f;
  S2[tid] = 0.f;
  __syncthreads();

  int osub  = wave & 3;      // which 4-outcap subtile of the 16
  int half  = wave >> 2;     // which half of the 16 hw-groups
  int col   = lane & 15;     // WMMA column
  int khalf = (lane >> 4) << 1;        // K = {0,1} or {2,3}
  int o_local = osub * 4 + (col >> 2);
  int o  = o0 + o_local;
  int pc = col & 3;
  int inst_a = col >> 2;     // A-matrix row -> instance
  int pr_a   = col & 3;
  int inst0  = (lane >> 4) << 1;       // D rows -> instance slots {0,1}/{2,3}

  float inv_rs = 1.f / (Rs[b * O_ + o] + EPSF);

  float acc1[8], acc2[8];
#pragma unroll
  for (int r = 0; r < 8; ++r) { acc1[r] = 0.f; acc2[r] = 0.f; }

  for (int i = 0; i < I_; ++i) {
    // B[k][col] = W[i, o(col), k, pc(col)]
    const float* wp = Wm + (size_t)(i * O_ + o) * 16 + pc;
    prefetch_g(Wm + (size_t)((((i + 1) & 31)) * O_ + o) * 16 + pc);
    v2f breg;
    breg.x = wp[khalf * 4];
    breg.y = wp[khalf * 4 + 4];
    for (int hh = 0; hh < 8; ++hh) {
      int hwg = half * 8 + hh;
      int hw  = hwg * 4 + inst_a;
      // A[m][k] = x[b, hw(inst), i, pr, k]
      const float* xp = x + ((size_t)(b * HW_ + hw) * I_ + i) * 16 + pr_a * 4;
      v2f areg;
      areg.x = xp[khalf];
      areg.y = xp[khalf + 1];
      v8f c8 = {};
      v8f d = wmma_f32_16x16x4(areg, breg, c8);
      // coeff for the two instance slots this lane's D rows cover
      int n0 = (hwg * 4 + inst0) * I_ + i;
      int n1 = n0 + I_;
      float c0 = R[(size_t)(b * O_ + o) * N_ + n0] * a[b * N_ + n0] * inv_rs;
      float c1 = R[(size_t)(b * O_ + o) * N_ + n1] * a[b * N_ + n1] * inv_rs;
#pragma unroll
      for (int r = 0; r < 4; ++r) {
        float v = d[r];
        acc1[r] += c0 * v;
        acc2[r] += c0 * v * v;
      }
#pragma unroll
      for (int r = 4; r < 8; ++r) {
        float v = d[r];
        acc1[r] += c1 * v;
        acc2[r] += c1 * v * v;
      }
    }
  }

  // Reduce instance slots: regs (pr, pr+4) then lane pair (l, l^16).
#pragma unroll
  for (int pr = 0; pr < 4; ++pr) {
    float s1 = acc1[pr] + acc1[pr + 4];
    float s2 = acc2[pr] + acc2[pr + 4];
    s1 += __shfl_xor(s1, 16, 32);
    s2 += __shfl_xor(s2, 16, 32);
    if (lane < 16) {
      atomicAdd(&S1[o_local * 16 + pr * 4 + pc], s1);  // ds_add_f32
      atomicAdd(&S2[o_local * 16 + pr * 4 + pc], s2);
    }
  }
  __syncthreads();

  // Finalize: one thread per (o_local, p).
  int ol = tid >> 4;
  int p  = tid & 15;
  int oo = o0 + ol;
  float rs_raw = Rs[b * O_ + oo];
  float cfrac  = rs_raw / (rs_raw + EPSF);
  float m1  = S1[ol * 16 + p];
  float s2v = S2[ol * 16 + p];
  float sig = s2v - m1 * m1 * (2.f - cfrac);
  sig = fmaxf(sig, 0.f) + EPSF;
  mu[(size_t)(b * O_ + oo) * 16 + p]    = m1;
  sigma[(size_t)(b * O_ + oo) * 16 + p] = sig;
  if (write_out) out_pose[(size_t)(b * O_ + oo) * 16 + p] = m1;
  float cost = (beta_u[oo] + 0.5f * logf(sig)) * rs_raw;
  cost += __shfl_xor(cost, 1, 32);
  cost += __shfl_xor(cost, 2, 32);
  cost += __shfl_xor(cost, 4, 32);
  cost += __shfl_xor(cost, 8, 32);
  if (p == 0) {
    float act = 1.f / (1.f + expf(-lam * (beta_a[oo] - cost)));
    aout[b * O_ + oo] = act;
    if (write_out) out_act[b * O_ + oo] = act;
  }
}

// ---------------------------------------------------------------------------
// E-step: per block = (b, group of 4 hw instances). For each i, recompute
// votes against all 64 output caps (16 WMMA subtiles split over 8 waves),
// form Gaussian log-probs, softmax over o, write R.
// Gaussian parameters (mu, 1/sigma, sum log sigma) are i-invariant and
// hoisted out of the streaming loop; softmax is fully parallel.
__global__ __launch_bounds__(256) void estep_kernel(
    const float* __restrict__ x,     // (B,HW,I,16)
    const float* __restrict__ Wm,    // (I,O,16)
    const float* __restrict__ mu,    // (B,O,16)
    const float* __restrict__ sigma, // (B,O,16)
    const float* __restrict__ aout,  // (B,O)
    float* __restrict__ R) {         // (B,O,N)
  __shared__ float LNP[4 * 64];
  __shared__ float MXw[8], SMw[8];
  const float ln2pi = 1.837877066409345f;

  int tid  = threadIdx.x;
  int lane = tid & 31;
  int wave = tid >> 5;
  int b    = blockIdx.x >> 4;
  int hwg  = blockIdx.x & 15;

  int col    = lane & 15;
  int khalf  = (lane >> 4) << 1;
  int inst_a = col >> 2;
  int pr_a   = col & 3;
  int pc     = col & 3;
  int instLo = (lane >> 4) << 1;

  // Per-lane output-cap indices for the wave's two o-subtiles.
  int o_s[2];
  o_s[0] = (wave * 2 + 0) * 4 + (col >> 2);
  o_s[1] = (wave * 2 + 1) * 4 + (col >> 2);

  // Hoisted Gaussian parameters: mv[s][r], inv-sigma[s][r], and
  // base[s] = sum_r (log sigma + ln2pi) for this lane's 4 pose rows.
  float mv[2][4], isv[2][4], base[2];
#pragma unroll
  for (int s = 0; s < 2; ++s) {
    base[s] = 4.f * ln2pi;
#pragma unroll
    for (int r = 0; r < 4; ++r) {
      int p = r * 4 + pc;
      float sv = sigma[(size_t)(b * O_ + o_s[s]) * 16 + p];
      mv[s][r]  = mu[(size_t)(b * O_ + o_s[s]) * 16 + p];
      isv[s][r] = 1.f / sv;
      base[s]  += logf(sv);
    }
  }

  // Per-thread softmax constants (inst = tid>>6, o = tid&63 are fixed).
  int instT = tid >> 6;
  int oT    = tid & 63;
  float logA = logf(aout[b * O_ + oT] + EPSF);
  int wpair  = wave & 6;   // base wave of this inst group's wave pair

  for (int i = 0; i < I_; ++i) {
    // A fragment is shared by both o-subtiles: load once per i.
    int hw = hwg * 4 + inst_a;
    const float* xp = x + ((size_t)(b * HW_ + hw) * I_ + i) * 16 + pr_a * 4;
    prefetch_g(x + ((size_t)(b * HW_ + hw) * I_ + ((i + 1) & 31)) * 16 + pr_a * 4);
    v2f areg;
    areg.x = xp[khalf];
    areg.y = xp[khalf + 1];

#pragma unroll
    for (int s = 0; s < 2; ++s) {
      int o = o_s[s];
      const float* wp = Wm + (size_t)(i * O_ + o) * 16 + pc;
      v2f breg;
      breg.x = wp[khalf * 4];
      breg.y = wp[khalf * 4 + 4];
      v8f c8 = {};
      v8f d = wmma_f32_16x16x4(areg, breg, c8);

      float u0 = base[s], u1 = base[s];
#pragma unroll
      for (int r = 0; r < 4; ++r) {
        float d0 = d[r] - mv[s][r];
        float d1 = d[r + 4] - mv[s][r];
        u0 += d0 * d0 * isv[s][r];
        u1 += d1 * d1 * isv[s][r];
      }
      // sum over the 4 pose columns (lanes within group of 4)
      u0 += __shfl_xor(u0, 1, 32);
      u0 += __shfl_xor(u0, 2, 32);
      u1 += __shfl_xor(u1, 1, 32);
      u1 += __shfl_xor(u1, 2, 32);
      if (pc == 0) {
        LNP[instLo * 64 + o]       = -0.5f * u0;
        LNP[(instLo + 1) * 64 + o] = -0.5f * u1;
      }
    }
    __syncthreads();

    // Parallel softmax over o (64) for each of 4 instances:
    // each thread owns one (inst, o) logit; reduce across the 2-wave group.
    float L = LNP[instT * 64 + oT] + logA;
    float mx = L;
#pragma unroll
    for (int off = 1; off < 32; off <<= 1)
      mx = fmaxf(mx, __shfl_xor(mx, off, 32));
    if (lane == 0) MXw[wave] = mx;
    __syncthreads();
    mx = fmaxf(MXw[wpair], MXw[wpair | 1]);
    float e = expf(L - mx);
    float sm = e;
#pragma unroll
    for (int off = 1; off < 32; off <<= 1)
      sm += __shfl_xor(sm, off, 32);
    if (lane == 0) SMw[wave] = sm;
    __syncthreads();
    sm = SMw[wpair] + SMw[wpair | 1];
    int n = (hwg * 4 + instT) * I_ + i;
    R[(size_t)(b * O_ + oT) * N_ + n] = e / sm;
    __syncthreads();   // LNP reused next i
  }
}

// ---------------------------------------------------------------------------
extern "C" void kernel_launch(void* const* d_in, const int* in_sizes, int n_in,
                              void* d_out, int out_size, void* d_ws,
                              size_t ws_size, hipStream_t stream) {
  const float* x      = (const float*)d_in[0];  // (B,Hh,Ww,I,4,4)
  const float* a      = (const float*)d_in[1];  // (B,Hh,Ww,I)
  const float* Wm     = (const float*)d_in[2];  // (I,O,4,4)
  const float* beta_u = (const float*)d_in[3];  // (1,O,1)
  const float* beta_a = (const float*)d_in[4];  // (1,O)

  float* out_pose = (float*)d_out;              // (B,O,16)
  float* out_act  = out_pose + B_ * O_ * P_;    // (B,O)

  float* ws    = (float*)d_ws;
  float* R     = ws;                                  // B*O*N = 4,194,304
  float* Rs    = R + (size_t)B_ * O_ * N_;            // 2048
  float* muW   = Rs + B_ * O_;                        // 32768
  float* sigW  = muW + B_ * O_ * P_;                  // 32768
  float* aoutW = sigW + B_ * O_ * P_;                 // 2048

  initR_kernel<<<(B_ * O_ * N_) / 256, 256, 0, stream>>>(R);

  for (int t = 0; t < 3; ++t) {
    float lam = 0.01f * (1.0f - powf(0.95f, (float)(t + 1)));
    int last  = (t == 2) ? 1 : 0;

    rs_kernel<<<B_ * O_, 256, 0, stream>>>(R, a, Rs);

    mstep_kernel<<<B_ * 4, 256, 0, stream>>>(x, a, Wm, beta_u, beta_a, R, Rs,
                                             muW, sigW, aoutW, out_pose,
                                             out_act, lam, last);

    if (t < 2)
      estep_kernel<<<B_ * 16, 256, 0, stream>>>(x, Wm, muW, sigW, aoutW, R);
  }
}